// BinomialTreePricer_47175920779512
// MI455X (gfx1250) — compile-verified
//
#include <hip/hip_runtime.h>
#include <math.h>

// CRR binomial American put, one option per wave32.
// 544-node padded lattice (32 lanes x 17 contiguous nodes in VGPRs).
// One ds_bpermute_b32 per backward step; 4 VALU ops per node per step:
//   t  = dq*V[k]
//   vn = fma(dp, V_up, t)
//   E  = fma(u, E, K*(1-u))        // exercise value K - S*u^(2j-s)
//   V  = max(vn, E)
// Nodes j > s never influence V[0], so padding beyond j=512 is provably inert.

#define NSTEPS 512
#define CPL 17          // V elements per lane: 32*17 = 544 >= 513
#define RRATE 0.03f
#define REFRESH 64      // steps between exp-based drift refreshes (512/64 = 8 blocks)

__global__ __launch_bounds__(256) void crr_american_put_kernel(
    const float* __restrict__ Sp, const float* __restrict__ Kp,
    const float* __restrict__ sigp, const float* __restrict__ Tp,
    float* __restrict__ out, int n)
{
    const int lane = (int)(threadIdx.x & 31u);
    const int wavesPerBlock = (int)(blockDim.x >> 5);
    const int w = (int)blockIdx.x * wavesPerBlock + (int)(threadIdx.x >> 5);
    if (w >= n) return;  // wave-uniform exit

    // gfx1250 prefetch path (global_prefetch_b8)
    __builtin_prefetch(Sp + w, 0, 0);
    __builtin_prefetch(Kp + w, 0, 0);
    __builtin_prefetch(sigp + w, 0, 0);
    __builtin_prefetch(Tp + w, 0, 0);

    const float S0 = Sp[w];
    const float K0 = Kp[w];
    const float sg = sigp[w];
    const float Tm = Tp[w];

    const float dt   = Tm * (1.0f / (float)NSTEPS);
    const float lnu  = sg * sqrtf(dt);          // ln(u)
    const float u    = expf(lnu);
    const float d    = 1.0f / u;
    const float erdt = expf(RRATE * dt);
    float p = (erdt - d) / (u - d + 1e-8f);
    p = fminf(fmaxf(p, 0.0f), 1.0f);
    const float disc = expf(-RRATE * dt);
    const float dp = disc * p;           // disc * p
    const float dq = disc * (1.0f - p);  // disc * (1-p)
    const float ec = K0 * (1.0f - u);    // E-recurrence constant: E <- fma(u, E, ec)

    // Lattice state: lane holds nodes j = jbase..jbase+16 (contiguous).
    const int jbase = lane * CPL;
    float V[CPL];   // option value
    float E[CPL];   // exercise value K - S*u^(2j - s), initialized at s = NSTEPS
    #pragma unroll
    for (int k = 0; k < CPL; ++k) {
        const float e = (float)(2 * (jbase + k) - NSTEPS);
        E[k] = K0 - S0 * expf(e * lnu);
        V[k] = fmaxf(E[k], 0.0f);
    }

    // bpermute byte index: read V[0] from lane (lane+1) & 31.
    const int bpIdx = (((lane + 1) & 31) << 2);

    int s = NSTEPS;  // current lattice exponent base (E = K - S*u^(2j-s))
    #pragma unroll 1
    for (int blk = 0; blk < NSTEPS / REFRESH; ++blk) {
        #pragma unroll 1
        for (int i = 0; i < REFRESH; ++i) {
            // neighbor lane's leading node (single cross-lane exchange per step)
            const float boundary = __int_as_float(
                __builtin_amdgcn_ds_bpermute(bpIdx, __float_as_int(V[0])));

            #pragma unroll
            for (int k = 0; k < CPL; ++k) {
                const float up = (k < CPL - 1) ? V[k + 1] : boundary;
                const float vn = fmaf(dp, up, dq * V[k]); // disc*(p*V_up+(1-p)*V)
                E[k] = fmaf(u, E[k], ec);                 // K - S*u^(2j-s)
                // vn >= 0 always, so the max(.,0) of the payoff is absorbed.
                V[k] = fmaxf(vn, E[k]);
            }
        }
        s -= REFRESH;
        // Refresh exercise values from exp to bound fma drift (8x per option).
        #pragma unroll
        for (int k = 0; k < CPL; ++k) {
            E[k] = K0 - S0 * expf((float)(2 * (jbase + k) - s) * lnu);
        }
    }

    if (lane == 0) out[w] = V[0];
}

extern "C" void kernel_launch(void* const* d_in, const int* in_sizes, int n_in,
                              void* d_out, int out_size, void* d_ws, size_t ws_size,
                              hipStream_t stream) {
    (void)n_in; (void)d_ws; (void)ws_size; (void)out_size;
    const float* S  = (const float*)d_in[0];
    const float* K  = (const float*)d_in[1];
    const float* sg = (const float*)d_in[2];
    const float* T  = (const float*)d_in[3];
    float* out = (float*)d_out;

    const int n = in_sizes[0];
    const int wavesPerBlock = 8;                 // 256 threads = 8 wave32
    const int blocks = (n + wavesPerBlock - 1) / wavesPerBlock;
    crr_american_put_kernel<<<blocks, 256, 0, stream>>>(S, K, sg, T, out, n);
}